// CTCLossLayer_4690104287651
// MI455X (gfx1250) — compile-verified
//
#include <hip/hip_runtime.h>
#include <cstdint>

// CTC loss for B=64, T=1000, V=512, L=256 (S = 2L+1 = 513), blank = 0.
// Two-pass design:
//   pass1: per-(b,t) log-softmax denominator + compact gather of 257 symbols
//   pass2: forward DP, 1 block per batch element, async LDS prefetch pipeline
// Fallback single fused kernel if workspace is too small (decided from ws_size,
// which is constant across calls -> deterministic).

namespace {
constexpr int   B    = 64;
constexpr int   T    = 1000;
constexpr int   V    = 512;
constexpr int   L    = 256;
constexpr int   S    = 2 * L + 1;   // 513 extended states
constexpr int   RS   = 264;         // compact row stride in floats (16B aligned)
constexpr float NEGV = -1e30f;      // matches reference NEG
}

// ---------------------------------------------------------------------------
// CDNA5 async global->LDS copy (gfx1250). VDST operand carries the LDS byte
// address; GVS addressing: saddr(64b SGPR pair) + per-lane 32b byte offset.
// Tracked by ASYNCcnt; we wait with exact counted s_wait_asynccnt.
// ---------------------------------------------------------------------------
#define S_WAIT_ASYNCCNT(N) asm volatile("s_wait_asynccnt " #N ::: "memory")

__device__ __forceinline__ void async_ld_f32(const float* sbase,
                                             unsigned     voff_bytes,
                                             const float* lds_dst) {
  // Low 32 bits of a flat pointer into the LDS aperture are the workgroup-
  // relative LDS byte address (ISA 10.2 aperture mapping).
  unsigned lds_off = (unsigned)(uintptr_t)lds_dst;
  asm volatile("global_load_async_to_lds_b32 %0, %1, %2"
               :
               : "v"(lds_off), "v"(voff_bytes), "s"(sbase)
               : "memory");
}

__device__ __forceinline__ float lse3(float a, float b, float c) {
  float m = fmaxf(a, fmaxf(b, c));
  return m + __logf(__expf(a - m) + __expf(b - m) + __expf(c - m));
}

// ---------------------------------------------------------------------------
// Pass 1: one block per (b,t) row. 256 threads, 2 logits each.
// Computes logZ = max + log(sum exp(x-max)) and writes the compact row:
//   lp[b,t,0]   = logits[b,t,0]      - logZ            (blank)
//   lp[b,t,k+1] = logits[b,t,lab[k]] - logZ, k=0..255  (labels; pad->blank col)
// ---------------------------------------------------------------------------
__global__ __launch_bounds__(256) void ctc_pass1(
    const float* __restrict__ logits, const int* __restrict__ labels,
    float* __restrict__ lp) {
  const int t   = blockIdx.x;
  const int b   = blockIdx.y;
  const int tid = threadIdx.x;

  __shared__ float row[V];
  __shared__ float wmax[8], wsum[8];
  __shared__ float s_logZ;

  const float* src = logits + ((size_t)b * T + t) * V;
  float x0 = src[tid];
  float x1 = src[tid + 256];
  row[tid]       = x0;
  row[tid + 256] = x1;

  // online logsumexp: per-thread pair, then wave32 shfl tree, then 8 partials
  float m  = fmaxf(x0, x1);
  float sm = __expf(x0 - m) + __expf(x1 - m);
#pragma unroll
  for (int off = 16; off >= 1; off >>= 1) {
    float m2 = __shfl_xor(m, off);
    float s2 = __shfl_xor(sm, off);
    float mn = fmaxf(m, m2);
    sm = sm * __expf(m - mn) + s2 * __expf(m2 - mn);
    m  = mn;
  }
  const int wid = tid >> 5, lane = tid & 31;
  if (lane == 0) { wmax[wid] = m; wsum[wid] = sm; }
  __syncthreads();
  if (tid == 0) {
    float M = wmax[0], SS = wsum[0];
#pragma unroll
    for (int i = 1; i < 8; ++i) {
      float m2 = wmax[i], s2 = wsum[i];
      float mn = fmaxf(M, m2);
      SS = SS * __expf(M - mn) + s2 * __expf(m2 - mn);
      M  = mn;
    }
    s_logZ = M + __logf(SS);
  }
  __syncthreads();
  const float logZ = s_logZ;

  float* dst = lp + ((size_t)b * T + t) * RS;
  for (int j = tid; j <= L; j += 256) {   // 257 outputs; tid0 writes j=0 and 256
    int sym = (j == 0) ? 0 : labels[b * L + (j - 1)];  // pad label 0 -> blank col
    dst[j] = row[sym] - logZ;
  }
}

// ---------------------------------------------------------------------------
// Pass 2: forward DP. 1 block per batch element, 512 threads (state s = tid;
// tid 0 also owns s = 512). Compact rows streamed via async-to-LDS, prefetch
// depth 3, 4 buffers; exactly one barrier per time step.
// ---------------------------------------------------------------------------
__global__ __launch_bounds__(512) void ctc_dp_pre(
    const float* __restrict__ lp, const int* __restrict__ labels,
    float* __restrict__ out) {
  const int b   = blockIdx.x;
  const int tid = threadIdx.x;

  __shared__ float buf[4][RS];
  __shared__ float alpha[2][S];
  __shared__ int   lab[L];
  __shared__ int   s_len;

  if (tid == 0) s_len = 0;
  __syncthreads();
  if (tid < L) {
    int v = labels[b * L + tid];
    lab[tid] = v;
    if (v != 0) atomicAdd(&s_len, 1);
  }
  __syncthreads();

  // Per-state constants. Compact-row index: even s -> 0 ; odd s=2k+1 -> k+1.
  int  idx0;
  bool skip0;
  if (tid & 1) {
    int k = tid >> 1;
    idx0  = k + 1;
    skip0 = (k >= 1) && (lab[k] != 0) && (lab[k] != lab[k - 1]);
  } else {
    idx0  = 0;
    skip0 = false;
  }

  const float* base = lp + (size_t)b * T * RS;

  // Prime the pipeline: rows 0..2. Only lanes covering the 257 columns load.
  if (tid <= L) {
#pragma unroll
    for (int r = 0; r < 3; ++r)
      async_ld_f32(base + (size_t)r * RS, (unsigned)(tid * 4), &buf[r][tid]);
  }

  int p = 0;
  for (int t = 0; t < T; ++t) {
    // Exact counted wait: rows t+1..min(t+2,T-1) may stay in flight.
    if (t + 2 <= T - 1)      S_WAIT_ASYNCCNT(2);
    else if (t + 1 <= T - 1) S_WAIT_ASYNCCNT(1);
    else                     S_WAIT_ASYNCCNT(0);
    __syncthreads();  // publishes row t AND alpha writes from step t-1

    const int cur = t & 3;
    if (t + 3 < T && tid <= L)  // refill buffer last read at step t-1 (safe)
      async_ld_f32(base + (size_t)(t + 3) * RS, (unsigned)(tid * 4),
                   &buf[(t + 3) & 3][tid]);

    const float lpv = buf[cur][idx0];
    if (t == 0) {
      alpha[0][tid] = (tid < 2) ? lpv : NEGV;
      if (tid == 0) alpha[0][S - 1] = NEGV;
    } else {
      const float* A  = alpha[p];
      float*       Nw = alpha[p ^ 1];
      float a  = A[tid];
      float a1 = (tid >= 1) ? A[tid - 1] : NEGV;
      float a2 = skip0 ? A[tid - 2] : NEGV;
      Nw[tid] = lse3(a, a1, a2) + lpv;
      if (tid == 0)  // s = 512: blank, no skip
        Nw[S - 1] = lse3(A[S - 1], A[S - 2], NEGV) + buf[cur][0];
      p ^= 1;
    }
  }
  __syncthreads();

  if (tid == 0) {
    int   len = s_len;
    int   eb  = 2 * len;
    int   el  = (2 * len - 1 > 0) ? (2 * len - 1) : 0;
    float v1 = alpha[p][eb], v2 = alpha[p][el];
    float mx = fmaxf(v1, v2);
    out[b] = -(mx + __logf(__expf(v1 - mx) + __expf(v2 - mx)));
  }
}

// ---------------------------------------------------------------------------
// Fallback: fused single kernel (used only if ws can't hold the compact lp).
// Same async pipeline but streams the full 512-wide logits row and does an
// online-logsumexp block reduction each step.
// ---------------------------------------------------------------------------
__global__ __launch_bounds__(512) void ctc_fused(
    const float* __restrict__ logits, const int* __restrict__ labels,
    float* __restrict__ out) {
  const int b   = blockIdx.x;
  const int tid = threadIdx.x;

  __shared__ float buf[4][V];
  __shared__ float alpha[2][S];
  __shared__ int   lab[L];
  __shared__ int   s_len;
  __shared__ float wmax[16], wsum[16];
  __shared__ float s_logZ;

  if (tid == 0) s_len = 0;
  __syncthreads();
  if (tid < L) {
    int v = labels[b * L + tid];
    lab[tid] = v;
    if (v != 0) atomicAdd(&s_len, 1);
  }
  __syncthreads();

  int  sym0;
  bool skip0;
  if (tid & 1) {
    int k = tid >> 1;
    sym0  = lab[k];
    skip0 = (k >= 1) && (sym0 != 0) && (sym0 != lab[k - 1]);
  } else {
    sym0  = 0;
    skip0 = false;
  }

  const float* base = logits + (size_t)b * T * V;
#pragma unroll
  for (int r = 0; r < 3; ++r)
    async_ld_f32(base + (size_t)r * V, (unsigned)(tid * 4), &buf[r][tid]);

  int p = 0;
  for (int t = 0; t < T; ++t) {
    if (t + 2 <= T - 1)      S_WAIT_ASYNCCNT(2);
    else if (t + 1 <= T - 1) S_WAIT_ASYNCCNT(1);
    else                     S_WAIT_ASYNCCNT(0);
    __syncthreads();

    const int cur = t & 3;
    if (t + 3 < T)
      async_ld_f32(base + (size_t)(t + 3) * V, (unsigned)(tid * 4),
                   &buf[(t + 3) & 3][tid]);

    // online logsumexp over the 512-wide row
    float x  = buf[cur][tid];
    float m  = x;
    float sm = 1.0f;
#pragma unroll
    for (int off = 16; off >= 1; off >>= 1) {
      float m2 = __shfl_xor(m, off);
      float s2 = __shfl_xor(sm, off);
      float mn = fmaxf(m, m2);
      sm = sm * __expf(m - mn) + s2 * __expf(m2 - mn);
      m  = mn;
    }
    const int wid = tid >> 5, lane = tid & 31;
    if (lane == 0) { wmax[wid] = m; wsum[wid] = sm; }
    __syncthreads();
    if (tid == 0) {
      float M = wmax[0], SS = wsum[0];
#pragma unroll
      for (int i = 1; i < 16; ++i) {
        float m2 = wmax[i], s2 = wsum[i];
        float mn = fmaxf(M, m2);
        SS = SS * __expf(M - mn) + s2 * __expf(m2 - mn);
        M  = mn;
      }
      s_logZ = M + __logf(SS);
    }
    __syncthreads();
    const float logZ = s_logZ;
    const float lpv  = buf[cur][sym0] - logZ;

    if (t == 0) {
      alpha[0][tid] = (tid < 2) ? lpv : NEGV;
      if (tid == 0) alpha[0][S - 1] = NEGV;
    } else {
      const float* A  = alpha[p];
      float*       Nw = alpha[p ^ 1];
      float a  = A[tid];
      float a1 = (tid >= 1) ? A[tid - 1] : NEGV;
      float a2 = skip0 ? A[tid - 2] : NEGV;
      Nw[tid] = lse3(a, a1, a2) + lpv;
      if (tid == 0)
        Nw[S - 1] = lse3(A[S - 1], A[S - 2], NEGV) + (buf[cur][0] - logZ);
      p ^= 1;
    }
  }
  __syncthreads();

  if (tid == 0) {
    int   len = s_len;
    int   eb  = 2 * len;
    int   el  = (2 * len - 1 > 0) ? (2 * len - 1) : 0;
    float v1 = alpha[p][eb], v2 = alpha[p][el];
    float mx = fmaxf(v1, v2);
    out[b] = -(mx + __logf(__expf(v1 - mx) + __expf(v2 - mx)));
  }
}

// ---------------------------------------------------------------------------
extern "C" void kernel_launch(void* const* d_in, const int* in_sizes, int n_in,
                              void* d_out, int out_size, void* d_ws,
                              size_t ws_size, hipStream_t stream) {
  (void)in_sizes; (void)n_in; (void)out_size;
  const float* logits = (const float*)d_in[0];
  const int*   labels = (const int*)d_in[1];  // harness ABI: integer -> int32
  float*       out    = (float*)d_out;

  const size_t need = (size_t)B * T * RS * sizeof(float);  // ~67.6 MB
  if (ws_size >= need) {
    float* lp = (float*)d_ws;
    dim3 g1(T, B);
    ctc_pass1<<<g1, 256, 0, stream>>>(logits, labels, lp);
    ctc_dp_pre<<<B, 512, 0, stream>>>(lp, labels, out);
  } else {
    ctc_fused<<<B, 512, 0, stream>>>(logits, labels, out);
  }
}